// MrcnnTarget_17789754541027
// MI455X (gfx1250) — compile-verified
//
#include <hip/hip_runtime.h>
#include <hip/hip_bf16.h>
#include <stdint.h>

// ---- problem constants (match reference) ----
#define TRAIN_ROIS 256
#define POS_QUOTA  25          // int(0.1 * 256)
#define NBOX_G     32          // gt boxes per image
#define GT_FLOATS  (8 * NBOX_G * 6)   // 1536 floats = 6144 bytes
#define POS_THR    0.5f
#define NEG_THR    0.02f
#define IOU_EPS    1e-8f

typedef unsigned int u32x4 __attribute__((ext_vector_type(4)));
typedef int          i32x4 __attribute__((ext_vector_type(4)));
typedef int          i32x8 __attribute__((ext_vector_type(8)));

// ---------------------------------------------------------------------------
// Phase 1: per-proposal IoU vs own-image GTs; emit pos/neg tie bitmasks.
// GT table staged into LDS via the Tensor Data Mover (one DMA per block).
// ---------------------------------------------------------------------------
__global__ __launch_bounds__(256) void mt_mask_kernel(
    const float* __restrict__ proposals,   // [P,6]
    const int*   __restrict__ batch,       // [P]
    const float* __restrict__ gt_boxes,    // [B,32,6] flat
    unsigned* __restrict__ posmask,        // [P]
    unsigned* __restrict__ negmask,        // [P]
    int P)
{
  __shared__ float sgt[GT_FLOATS];

#if __has_builtin(__builtin_amdgcn_tensor_load_to_lds)
  if (threadIdx.x < 32) {          // wave 0 only issues the TDM op
    const uint64_t ga      = (uint64_t)(uintptr_t)gt_boxes;
    const uint32_t lds_off = (uint32_t)(uintptr_t)&sgt[0];   // low 32b of LDS flat ptr = LDS offset
    u32x4 g0;
    g0[0] = 1u;                                   // count=1 valid, user mode, no gather
    g0[1] = lds_off;                              // lds_addr (bytes)
    g0[2] = (uint32_t)ga;                         // global_addr[31:0]
    g0[3] = (uint32_t)((ga >> 32) & 0x01FFFFFFull) | (2u << 30);  // addr[56:32] | type=2
    i32x8 g1;
    g1[0] = (int)(2u << 16);                      // data_size = 4 bytes
    g1[1] = (int)((GT_FLOATS & 0xFFFF) << 16);    // tensor_dim0 low16
    g1[2] = (int)(((unsigned)GT_FLOATS >> 16) | (1u << 16)); // dim0 hi16 | tensor_dim1=1
    g1[3] = (int)((GT_FLOATS & 0xFFFF) << 16);    // tile_dim0 = 1536
    g1[4] = 1;                                    // tile_dim1=1, tile_dim2=0
    g1[5] = GT_FLOATS;                            // tensor_dim0_stride
    g1[6] = 0; g1[7] = 0;
    i32x4 z4 = {0, 0, 0, 0};
#if __clang_major__ >= 23
    i32x8 z8 = {0, 0, 0, 0, 0, 0, 0, 0};
    __builtin_amdgcn_tensor_load_to_lds(g0, g1, z4, z4, z8, 0);
#else
    __builtin_amdgcn_tensor_load_to_lds(g0, g1, z4, z4, 0);
#endif
    __builtin_amdgcn_s_wait_tensorcnt(0);
  }
#else
  for (int k = threadIdx.x; k < GT_FLOATS; k += blockDim.x) sgt[k] = gt_boxes[k];
#endif
  __syncthreads();

  const int p = blockIdx.x * blockDim.x + threadIdx.x;
  if (p >= P) return;
  __builtin_prefetch(&proposals[(size_t)p * 6], 0, 0);   // global_prefetch_b8

  const int img = batch[p];
  float b[6];
#pragma unroll
  for (int k = 0; k < 6; ++k) b[k] = proposals[(size_t)p * 6 + k];
  const float vol_p = (b[3] - b[0]) * (b[4] - b[1]) * (b[5] - b[2]);

  const float* q0 = &sgt[img * NBOX_G * 6];
  float iou[NBOX_G];
  float m = -1.0f;
#pragma unroll
  for (int g = 0; g < NBOX_G; ++g) {
    const float* q = q0 + g * 6;
    const float d0 = fminf(b[3], q[3]) - fmaxf(b[0], q[0]);
    const float d1 = fminf(b[4], q[4]) - fmaxf(b[1], q[1]);
    const float d2 = fminf(b[5], q[5]) - fmaxf(b[2], q[2]);
    const float inter = fmaxf(d0, 0.f) * fmaxf(d1, 0.f) * fmaxf(d2, 0.f);
    const float vg = (q[3] - q[0]) * (q[4] - q[1]) * (q[5] - q[2]);
    iou[g] = inter / (vg + vol_p - inter + IOU_EPS);
    m = fmaxf(m, iou[g]);
  }
  unsigned pm = 0u, nm = 0u;
#pragma unroll
  for (int g = 0; g < NBOX_G; ++g) {
    if (iou[g] == m) {                 // exact tie semantics as jnp (iou == roi_max)
      if (m >= POS_THR)      pm |= (1u << g);
      else if (m < NEG_THR)  nm |= (1u << g);   // in-image iou >= 0 always
    }
  }
  posmask[p] = pm;
  negmask[p] = nm;
}

// ---------------------------------------------------------------------------
// Phase 2: ordered first-k selection per image (gt-major, proposal-minor),
// wave32 ballot + block prefix reproduces the stable argsort-of-mask order.
// ---------------------------------------------------------------------------
__global__ __launch_bounds__(256) void mt_select_kernel(
    const int*      __restrict__ batch,
    const unsigned* __restrict__ posmask,
    const unsigned* __restrict__ negmask,
    int P,
    int* __restrict__ sel_g, int* __restrict__ sel_p, int* __restrict__ sel_kind)
{
  const int img  = blockIdx.x;
  const int t    = threadIdx.x;
  const int lane = t & 31;
  const int wv   = t >> 5;

  __shared__ int count, done, posnum;
  __shared__ int woff[8];

  // init all 256 slots invalid
  sel_kind[img * TRAIN_ROIS + t] = 0;
  sel_g[img * TRAIN_ROIS + t]    = 0;
  sel_p[img * TRAIN_ROIS + t]    = 0;
  if (t == 0) { count = 0; done = 0; posnum = 0; }
  __syncthreads();

  for (int pass = 0; pass < 2; ++pass) {
    __syncthreads();
    const int sbase = pass ? posnum : 0;
    const int quota = pass ? (TRAIN_ROIS - posnum) : POS_QUOTA;
    const int kind  = pass ? -1 : 1;
    const unsigned* __restrict__ mask = pass ? negmask : posmask;

    bool stop = (quota <= 0);
    for (int g = 0; g < NBOX_G && !stop; ++g) {
      for (int base = 0; base < P && !stop; base += 256) {
        const int p = base + t;
        const bool flag = (p < P) && (batch[p] == img) && ((mask[p] >> g) & 1u);
        const unsigned bal = (unsigned)__ballot(flag);     // wave32 mask
        if (lane == 0) woff[wv] = __popc(bal);
        __syncthreads();
        if (t == 0) {
          int run = count;
          for (int w = 0; w < 8; ++w) { const int c = woff[w]; woff[w] = run; run += c; }
          count = run;
          if (run >= quota) done = 1;
        }
        __syncthreads();
        if (flag) {
          const int idx = woff[wv] + __popc(bal & ((1u << lane) - 1u));
          if (idx < quota) {
            sel_kind[img * TRAIN_ROIS + sbase + idx] = kind;
            sel_g[img * TRAIN_ROIS + sbase + idx]    = g;
            sel_p[img * TRAIN_ROIS + sbase + idx]    = p;
          }
        }
        __syncthreads();
        stop = (done != 0);
      }
    }
    __syncthreads();
    if (t == 0) {
      int c = count; if (c > quota) c = quota;
      if (pass == 0) posnum = c;
      count = 0; done = 0;
    }
  }
}

// ---------------------------------------------------------------------------
// Phase 3: materialize rois / deltas / labels / tag / ridx (concatenated f32).
// ---------------------------------------------------------------------------
__global__ __launch_bounds__(256) void mt_emit_kernel(
    const float* __restrict__ proposals,
    const float* __restrict__ gt_boxes,
    const int*   __restrict__ gt_labels,
    const int* __restrict__ sel_g, const int* __restrict__ sel_p,
    const int* __restrict__ sel_kind,
    float* __restrict__ out, int N)
{
  const int s = blockIdx.x * blockDim.x + threadIdx.x;
  if (s >= N) return;
  const int img = s / TRAIN_ROIS;

  float* rois   = out;
  float* deltas = out + (size_t)N * 6;
  float* labs   = out + (size_t)N * 12;
  float* tags   = labs + N;
  float* ridx   = tags + N;

  const int kind = sel_kind[s];
  if (kind == 0) {
#pragma unroll
    for (int k = 0; k < 6; ++k) { rois[s * 6 + k] = 0.f; deltas[s * 6 + k] = 0.f; }
    labs[s] = 0.f; tags[s] = 0.f; ridx[s] = -1.f;
    return;
  }
  const int p = sel_p[s], g = sel_g[s];
  const float* r = proposals + (size_t)p * 6;
  const float* q = gt_boxes + (size_t)(img * NBOX_G + g) * 6;
#pragma unroll
  for (int k = 0; k < 6; ++k) rois[s * 6 + k] = r[k];
#pragma unroll
  for (int a = 0; a < 3; ++a) {
    const float sz = r[a + 3] - r[a];
    const float c  = (r[a] + r[a + 3]) * 0.5f;
    const float gs = q[a + 3] - q[a];
    const float gc = (q[a] + q[a + 3]) * 0.5f;
    deltas[s * 6 + a]     = (gc - c) / sz;
    deltas[s * 6 + 3 + a] = logf(gs / sz);
  }
  labs[s] = (kind == 1) ? (float)gt_labels[img * NBOX_G + g] : 0.f;
  tags[s] = (kind == 1) ? 1.f : -1.f;
  ridx[s] = (float)img;
}

// ---------------------------------------------------------------------------
extern "C" void kernel_launch(void* const* d_in, const int* in_sizes, int n_in,
                              void* d_out, int out_size, void* d_ws, size_t ws_size,
                              hipStream_t stream) {
  const float* proposals = (const float*)d_in[0];
  const int*   batch     = (const int*)d_in[1];
  const float* gt_boxes  = (const float*)d_in[2];
  const int*   gt_labels = (const int*)d_in[3];

  const int P = in_sizes[0] / 6;                 // 65536
  const int B = (in_sizes[2] / 6) / NBOX_G;      // 8
  const int N = B * TRAIN_ROIS;                  // 2048

  unsigned* posmask = (unsigned*)d_ws;
  unsigned* negmask = posmask + P;
  int* sel_g    = (int*)(negmask + P);
  int* sel_p    = sel_g + N;
  int* sel_kind = sel_p + N;

  mt_mask_kernel<<<(P + 255) / 256, 256, 0, stream>>>(
      proposals, batch, gt_boxes, posmask, negmask, P);
  mt_select_kernel<<<B, 256, 0, stream>>>(
      batch, posmask, negmask, P, sel_g, sel_p, sel_kind);
  mt_emit_kernel<<<(N + 255) / 256, 256, 0, stream>>>(
      proposals, gt_boxes, gt_labels, sel_g, sel_p, sel_kind, (float*)d_out, N);
}